// QBuilder_21345987461695
// MI455X (gfx1250) — compile-verified
//
#include <hip/hip_runtime.h>
#include <hip/hip_bf16.h>

#define EPS 1e-8f
#define LOG2E 1.44269504088896340736f

typedef __attribute__((ext_vector_type(2))) float v2f;
typedef __attribute__((ext_vector_type(8))) float v8f;

// ---------------------------------------------------------------------------
// Deterministic 256-thread block reduction (shared-mem tree, fixed order).
// ---------------------------------------------------------------------------
__device__ __forceinline__ float blockReduce256(float val) {
  __shared__ float sh[256];
  int t = threadIdx.x;
  sh[t] = val;
  __syncthreads();
#pragma unroll
  for (int off = 128; off > 0; off >>= 1) {
    if (t < off) sh[t] += sh[t + off];
    __syncthreads();
  }
  float res = sh[0];
  __syncthreads();  // allow safe reuse of sh on subsequent calls
  return res;
}

// ---------------------------------------------------------------------------
// q = X @ W + b   with V_WMMA_F32_16X16X4_F32.
// X: [N,K] row-major, W: [K,2] row-major, q: [N,2]. One wave (32 threads)
// computes a 16-row x 16-col tile (cols 2..15 are zero-masked B columns).
// A (16x4 f32): lanes 0-15 hold K={k0,k0+1}, lanes 16-31 hold K={k0+2,k0+3}.
// B (4x16 f32): VGPR0 = rows K=k0 (lanes 0-15) / K=k0+2 (lanes 16-31), etc.
// D (16x16 f32): VGPR i -> M = (lane>=16 ? 8 : 0) + i, N = lane & 15.
// B loads are branch-free: column index clamped to {0,1} (always a valid
// address) and masked by a VALU multiply -> no saveexec per K-step.
// ---------------------------------------------------------------------------
__global__ __launch_bounds__(32) void k_qgemm(const float* __restrict__ X,
                                              const float* __restrict__ W,
                                              const float* __restrict__ bias,
                                              float* __restrict__ q, int K) {
  const int lane    = threadIdx.x;          // 0..31
  const int rowBase = blockIdx.x * 16;
  const int rA      = rowBase + (lane & 15);
  const int kOff    = (lane >> 4) * 2;      // 0 or 2
  const int nCol    = lane & 15;

  const float* __restrict__ Xrow = X + (size_t)rA * K + kOff;
  const int   wcol  = nCol & 1;             // clamped, always in-bounds
  const float wmask = (nCol < 2) ? 1.0f : 0.0f;

  v8f acc = {};
  for (int k0 = 0; k0 < K; k0 += 4) {
    v2f a, b;
    a.x = Xrow[k0];
    a.y = Xrow[k0 + 1];
    b.x = W[(k0 + kOff) * 2 + wcol] * wmask;
    b.y = W[(k0 + kOff + 1) * 2 + wcol] * wmask;
    acc = __builtin_amdgcn_wmma_f32_16x16x4_f32(
        /*neg_a=*/false, a, /*neg_b=*/false, b,
        /*c_mod=*/(short)0, acc, /*reuse_a=*/false, /*reuse_b=*/false);
  }

  if (nCol < 2) {
    const float bv    = bias[nCol];
    const int   mHigh = (lane >> 4) * 8;
#pragma unroll
    for (int i = 0; i < 8; ++i) {
      q[(rowBase + mHigh + i) * 2 + nCol] = acc[i] + bv;
    }
  }
}

// ---------------------------------------------------------------------------
// Per-class (column) standardization: o = (q - mean)/(std_ddof1 + EPS) * scale.
// scale = LOG2E for the u-side (so every later exp(u*v) becomes exp2(u2*v)),
// scale = 1 for the v-side. One block per class c in {0,1}.
// ---------------------------------------------------------------------------
__global__ __launch_bounds__(256) void k_standardize(const float* __restrict__ q,
                                                     float* __restrict__ o,
                                                     float scale, int N) {
  const int c = blockIdx.x;
  float sum = 0.0f;
  for (int i = threadIdx.x; i < N; i += 256) sum += q[i * 2 + c];
  const float mean = blockReduce256(sum) / (float)N;

  float sq = 0.0f;
  for (int i = threadIdx.x; i < N; i += 256) {
    const float d = q[i * 2 + c] - mean;
    sq += d * d;
  }
  const float var = blockReduce256(sq) / (float)(N - 1);
  const float inv = scale / (sqrtf(var) + EPS);
  for (int i = threadIdx.x; i < N; i += 256)
    o[i * 2 + c] = (q[i * 2 + c] - mean) * inv;
}

// ---------------------------------------------------------------------------
// Per-row partial of S = sum_{a,b,c} max(exp2(u2_c[a]*v_c[b]), EPS).
// ---------------------------------------------------------------------------
__global__ __launch_bounds__(256) void k_rowS(const float* __restrict__ u2,
                                              const float* __restrict__ v,
                                              float* __restrict__ partial, int N) {
  const int   a  = blockIdx.x;
  const float u0 = u2[a * 2 + 0], u1 = u2[a * 2 + 1];
  float acc = 0.0f;
  for (int b = threadIdx.x; b < N; b += 256) {
    acc += fmaxf(__builtin_amdgcn_exp2f(u0 * v[b * 2 + 0]), EPS) +
           fmaxf(__builtin_amdgcn_exp2f(u1 * v[b * 2 + 1]), EPS);
  }
  const float tot = blockReduce256(acc);
  if (threadIdx.x == 0) partial[a] = tot;
}

// Deterministic final reduce of partials; init r = 1/S, s = 1 (interleaved [N][2]).
__global__ __launch_bounds__(256) void k_initRS(const float* __restrict__ partial,
                                                float* __restrict__ r,
                                                float* __restrict__ s, int N) {
  float acc = 0.0f;
  for (int i = threadIdx.x; i < N; i += 256) acc += partial[i];
  const float S     = blockReduce256(acc);
  const float rinit = 1.0f / S;
  for (int i = threadIdx.x; i < 2 * N; i += 256) {
    r[i] = rinit;
    s[i] = 1.0f;
  }
}

// ---------------------------------------------------------------------------
// Sinkhorn column step: s'_b = s_b * m2_b / clip(s_b * sum_a r_a 2^{u2_a v_b}, EPS)
// One block per (b, c). Reads r/u2 (fixed this step), updates only s[b*2+c].
// r,s interleaved [N][2].
// ---------------------------------------------------------------------------
__global__ __launch_bounds__(256) void k_update_s(const float* __restrict__ u2,
                                                  const float* __restrict__ v,
                                                  const float* __restrict__ r,
                                                  float* __restrict__ s,
                                                  const float* __restrict__ p2,
                                                  float invN, int N) {
  const int   b  = blockIdx.x;
  const int   c  = blockIdx.y;
  const float vb = v[b * 2 + c];
  float acc = 0.0f;
  for (int a = threadIdx.x; a < N; a += 256)
    acc += r[a * 2 + c] * __builtin_amdgcn_exp2f(u2[a * 2 + c] * vb);
  const float t = blockReduce256(acc);
  if (threadIdx.x == 0) {
    const float sb = s[b * 2 + c];
    const float m2 = p2[b * 2 + c] * invN;
    s[b * 2 + c] = sb * m2 / fmaxf(sb * t, EPS);
  }
}

// Sinkhorn row step: r'_a = r_a * m1_a / clip(r_a * sum_b 2^{u2_a v_b} s_b, EPS)
__global__ __launch_bounds__(256) void k_update_r(const float* __restrict__ u2,
                                                  const float* __restrict__ v,
                                                  float* __restrict__ r,
                                                  const float* __restrict__ s,
                                                  const float* __restrict__ p1,
                                                  float invN, int N) {
  const int   a  = blockIdx.x;
  const int   c  = blockIdx.y;
  const float ua = u2[a * 2 + c];
  float acc = 0.0f;
  for (int b = threadIdx.x; b < N; b += 256)
    acc += s[b * 2 + c] * __builtin_amdgcn_exp2f(ua * v[b * 2 + c]);
  const float t = blockReduce256(acc);
  if (threadIdx.x == 0) {
    const float ra = r[a * 2 + c];
    const float m1 = p1[a * 2 + c] * invN;
    r[a * 2 + c] = ra * m1 / fmaxf(ra * t, EPS);
  }
}

// ---------------------------------------------------------------------------
// Final materialization: Q_c = clip(r_c s_c 2^{u2 v}, EPS); class renorm
// (g_c = clip(Q_c/T, EPS, 1); out_c = g_c/(g0+g1) * T).
// Each thread handles a pair of b's -> b128 loads of v/s, b128 output stores.
// ---------------------------------------------------------------------------
__global__ __launch_bounds__(256) void k_writeQ(const float* __restrict__ u2,
                                                const float* __restrict__ v,
                                                const float* __restrict__ r,
                                                const float* __restrict__ s,
                                                float* __restrict__ out, int N) {
  const int   a  = blockIdx.x;
  const float u0 = u2[a * 2 + 0], u1 = u2[a * 2 + 1];
  const float r0 = r[a * 2 + 0],  r1 = r[a * 2 + 1];
  const float4* __restrict__ V4 = (const float4*)v;   // {v[b][0],v[b][1],v[b+1][0],v[b+1][1]}
  const float4* __restrict__ S4 = (const float4*)s;
  float4* __restrict__ O4 = (float4*)out;

  for (int b = threadIdx.x * 2; b < N; b += 512) {
    const float4 vv = V4[b >> 1];
    const float4 ss = S4[b >> 1];

    // pair element 0 (b)
    float q0 = fmaxf(r0 * ss.x * __builtin_amdgcn_exp2f(u0 * vv.x), EPS);
    float q1 = fmaxf(r1 * ss.y * __builtin_amdgcn_exp2f(u1 * vv.y), EPS);
    float T  = q0 + q1;
    float g0 = fminf(fmaxf(q0 / T, EPS), 1.0f);
    float g1 = fminf(fmaxf(q1 / T, EPS), 1.0f);
    float k  = T / (g0 + g1);

    // pair element 1 (b+1)
    float q2 = fmaxf(r0 * ss.z * __builtin_amdgcn_exp2f(u0 * vv.z), EPS);
    float q3 = fmaxf(r1 * ss.w * __builtin_amdgcn_exp2f(u1 * vv.w), EPS);
    float T2 = q2 + q3;
    float g2 = fminf(fmaxf(q2 / T2, EPS), 1.0f);
    float g3 = fminf(fmaxf(q3 / T2, EPS), 1.0f);
    float k2 = T2 / (g2 + g3);

    float4 res;
    res.x = g0 * k;
    res.y = g1 * k;
    res.z = g2 * k2;
    res.w = g3 * k2;
    O4[((size_t)a * N + b) >> 1] = res;   // a*N+b is even -> aligned b128
  }
}

// ---------------------------------------------------------------------------
extern "C" void kernel_launch(void* const* d_in, const int* in_sizes, int n_in,
                              void* d_out, int out_size, void* d_ws, size_t ws_size,
                              hipStream_t stream) {
  const float* x1 = (const float*)d_in[0];
  const float* x2 = (const float*)d_in[1];
  const float* p1 = (const float*)d_in[2];   // p(y|x1), [N,2]
  const float* p2 = (const float*)d_in[3];   // p(y|x2), [N,2]
  const float* W1 = (const float*)d_in[4];
  const float* b1 = (const float*)d_in[5];
  const float* W2 = (const float*)d_in[6];
  const float* b2 = (const float*)d_in[7];
  float*       Q  = (float*)d_out;

  const int N  = in_sizes[2] / 2;            // 2048
  const int D1 = in_sizes[0] / N;            // 256
  const int D2 = in_sizes[1] / N;            // 256
  const float invN = 1.0f / (float)N;        // p_x

  float* ws      = (float*)d_ws;
  float* q1      = ws;                        // [N,2]
  float* q2      = ws + 2 * N;                // [N,2]
  float* u2      = ws + 4 * N;                // [N,2] standardized q1 * log2(e)
  float* v       = ws + 6 * N;                // [N,2] standardized q2
  float* partial = ws + 8 * N;                // [N]
  float* r       = ws + 9 * N;                // [N,2] interleaved
  float* s       = ws + 11 * N;               // [N,2] interleaved

  // Projections via fp32 WMMA (16x16x4).
  k_qgemm<<<N / 16, 32, 0, stream>>>(x1, W1, b1, q1, D1);
  k_qgemm<<<N / 16, 32, 0, stream>>>(x2, W2, b2, q2, D2);

  // Per-class standardization (u-side pre-scaled by log2e for exp2 kernels).
  k_standardize<<<2, 256, 0, stream>>>(q1, u2, LOG2E, N);
  k_standardize<<<2, 256, 0, stream>>>(q2, v, 1.0f, N);

  // Global normalizer S (deterministic two-stage reduce), init r = 1/S, s = 1.
  k_rowS<<<N, 256, 0, stream>>>(u2, v, partial, N);
  k_initRS<<<1, 256, 0, stream>>>(partial, r, s, N);

  // 8 Sinkhorn iterations on the scaling vectors only (E recomputed on the fly).
  for (int it = 0; it < 8; ++it) {
    k_update_s<<<dim3(N, 2), 256, 0, stream>>>(u2, v, r, s, p2, invN, N);
    k_update_r<<<dim3(N, 2), 256, 0, stream>>>(u2, v, r, s, p1, invN, N);
  }

  // Single full-tensor pass: clip + class renorm + write [N,N,2].
  k_writeQ<<<N, 256, 0, stream>>>(u2, v, r, s, Q, N);
}